// StyleGANLayerEpilogue_29540785062600
// MI455X (gfx1250) — compile-verified
//
#include <hip/hip_runtime.h>

// ---------------------------------------------------------------------------
// StyleGAN layer epilogue for MI455X (gfx1250, wave32, WMMA).
// Main GEMM (32768x512x512 fp32) via V_WMMA_F32_16X16X4_F32 (native fp32 path,
// matches fp32 reference precision). Deterministic 3-kernel pipeline, no
// atomics, activations staged in d_out between kernels 1 and 3.
// B tile stored transposed in LDS so both A and B fragments are contiguous
// 8B-aligned pairs -> pure ds_load_b64 / ds_load_2addr_b64 + wmma inner loop.
// ---------------------------------------------------------------------------

typedef float v2f __attribute__((ext_vector_type(2)));
typedef float v8f __attribute__((ext_vector_type(8)));

#define B_      8
#define HW      4096        // 64*64 spatial per batch
#define C_      512         // K (input channels)
#define NF      512         // N (output features)
#define MTOT    32768       // B_ * HW
#define BM      128
#define BN      64
#define BK      32
#define MT_PER_B 32         // HW / BM
#define LDA     36          // padded LDS row stride for A  [m][k] (floats)
#define LDBT    36          // padded LDS row stride for Bt [n][k] (floats)
#define LRELU   0.2f

// ---------------------------------------------------------------------------
// Kernel 1: tiled fp32 WMMA GEMM + noise/bias/lrelu + per-block column stats.
// Grid: (MTOT/BM, NF/BN), Block: 256 threads = 8 waves.
// Wave w: wm = w&3 -> 32-row sub-tile, wn = w>>2 -> 32-col sub-tile.
// ---------------------------------------------------------------------------
__global__ __launch_bounds__(256) void k1_gemm_act_stats(
    const float* __restrict__ x,        // [MTOT, C_]
    const float* __restrict__ w,        // [C_, NF]
    const float* __restrict__ noise_map,// [MTOT]
    const float* __restrict__ noise_w,  // [NF]
    const float* __restrict__ bias,     // [NF]
    float* __restrict__ out,            // [MTOT, NF] (activations)
    float* __restrict__ part)           // [B_, MT_PER_B, NF, 2] partial sums
{
    __shared__ float As [BM * LDA];     // row-major  [m][k]
    __shared__ float Bst[BN * LDBT];    // transposed [n][k]
    __shared__ float sNoise[BM];
    __shared__ float sRed[8 * 32 * 4];  // [wave][lane][ni*2 + stat]

    const int tid  = threadIdx.x;
    const int lane = tid & 31;
    const int wave = tid >> 5;
    const int wm   = wave & 3;
    const int wn   = wave >> 2;
    const int colL = lane & 15;         // N (or M) index inside a 16-tile
    const int half = lane >> 4;         // lane half selects K pair / M+8

    const int mbase = blockIdx.x * BM;
    const int nbase = blockIdx.y * BN;
    const int b     = mbase / HW;
    const int mt    = (mbase % HW) / BM;

    if (tid < BM) sNoise[tid] = noise_map[mbase + tid];

    float nwv[2], bsv[2];
#pragma unroll
    for (int ni = 0; ni < 2; ++ni) {
        const int f = nbase + wn * 32 + ni * 16 + colL;
        nwv[ni] = noise_w[f];
        bsv[ni] = bias[f];
    }

    v8f acc[2][2] = {};

    for (int kb = 0; kb < C_; kb += BK) {
        // ---- stage A tile: 128x32, row-major (4 rounds of float4/thread) ----
        {
            const int r = tid >> 3;          // 0..31
            const int c = (tid & 7) << 2;    // 0..28
#pragma unroll
            for (int it = 0; it < 4; ++it) {
                const int row = r + it * 32;
                const float* gp = x + (size_t)(mbase + row) * C_ + kb + c;
                if (kb + BK < C_) __builtin_prefetch(gp + BK, 0, 0);
                *(float4*)(&As[row * LDA + c]) = *(const float4*)gp;
            }
        }
        // ---- stage B tile: 32x64 -> transposed [n][k] (2 rounds/thread) ----
        {
            const int r = tid >> 4;          // k row 0..15
            const int c = (tid & 15) << 2;   // n 0..60
#pragma unroll
            for (int it = 0; it < 2; ++it) {
                const int krow = r + it * 16;
                const float4 v = *(const float4*)(w + (size_t)(kb + krow) * NF + nbase + c);
                Bst[(c + 0) * LDBT + krow] = v.x;
                Bst[(c + 1) * LDBT + krow] = v.y;
                Bst[(c + 2) * LDBT + krow] = v.z;
                Bst[(c + 3) * LDBT + krow] = v.w;
            }
        }
        __syncthreads();

        // ---- 8 K-steps of 4: V_WMMA_F32_16X16X4_F32 ----
#pragma unroll
        for (int kr = 0; kr < BK; kr += 4) {
            const int ka = kr + half * 2;    // f32 WMMA layout: lanes 0-15 hold
                                             // K=kr..kr+1, lanes 16-31 K=kr+2..kr+3
            v2f a[2], bf[2];
#pragma unroll
            for (int mi = 0; mi < 2; ++mi) {
                const float* p = &As[(wm * 32 + mi * 16 + colL) * LDA + ka];
                a[mi].x = p[0];
                a[mi].y = p[1];
            }
#pragma unroll
            for (int ni = 0; ni < 2; ++ni) {
                const float* p = &Bst[(wn * 32 + ni * 16 + colL) * LDBT + ka];
                bf[ni].x = p[0];
                bf[ni].y = p[1];
            }
#pragma unroll
            for (int mi = 0; mi < 2; ++mi)
#pragma unroll
                for (int ni = 0; ni < 2; ++ni)
                    acc[mi][ni] = __builtin_amdgcn_wmma_f32_16x16x4_f32(
                        false, a[mi], false, bf[ni], (short)0, acc[mi][ni],
                        false, false);
        }
        __syncthreads();
    }

    // ---- epilogue: noise + bias + lrelu, store, per-lane column stats ----
    float cs[2] = {0.f, 0.f}, cq[2] = {0.f, 0.f};
#pragma unroll
    for (int mi = 0; mi < 2; ++mi) {
#pragma unroll
        for (int r = 0; r < 8; ++r) {
            const int mloc = wm * 32 + mi * 16 + half * 8 + r;  // C/D: M = r + half*8
            const float nz = sNoise[mloc];
            const size_t rowoff = (size_t)(mbase + mloc) * NF;
#pragma unroll
            for (int ni = 0; ni < 2; ++ni) {
                float v = acc[mi][ni][r] + nwv[ni] * nz + bsv[ni];
                v = (v > 0.f) ? v : LRELU * v;
                out[rowoff + nbase + wn * 32 + ni * 16 + colL] = v;
                cs[ni] += v;
                cq[ni] += v * v;
            }
        }
    }
#pragma unroll
    for (int ni = 0; ni < 2; ++ni) {
        sRed[(wave * 32 + lane) * 4 + ni * 2 + 0] = cs[ni];
        sRed[(wave * 32 + lane) * 4 + ni * 2 + 1] = cq[ni];
    }
    __syncthreads();

    // deterministic fixed-order reduction: 8 contributions per column
    if (tid < BN) {
        const int c   = tid;
        const int wn2 = c >> 5;
        const int ni  = (c >> 4) & 1;
        const int cc  = c & 15;
        float s = 0.f, q = 0.f;
#pragma unroll
        for (int wmi = 0; wmi < 4; ++wmi) {
            const int wv = wn2 * 4 + wmi;
#pragma unroll
            for (int h = 0; h < 2; ++h) {
                const int ln = cc + h * 16;
                s += sRed[(wv * 32 + ln) * 4 + ni * 2 + 0];
                q += sRed[(wv * 32 + ln) * 4 + ni * 2 + 1];
            }
        }
        const size_t pidx = (((size_t)b * MT_PER_B + mt) * NF + (nbase + c)) * 2;
        part[pidx]     = s;
        part[pidx + 1] = q;
    }
}

// ---------------------------------------------------------------------------
// Kernel 2: finalize stats + style GEMM -> per-(b,f) scale/shift.
// out = (y-mu)*rsig*(1+gamma)+beta = y*s + t,  s = rsig*(1+gamma), t = beta-mu*s
// ---------------------------------------------------------------------------
__global__ __launch_bounds__(256) void k2_stats_style(
    const float* __restrict__ part,
    const float* __restrict__ dlat,     // [B_, C_]
    const float* __restrict__ dw,       // [C_, 2*NF]
    const float* __restrict__ db,       // [2*NF]
    float* __restrict__ sBuf,           // [B_*NF]
    float* __restrict__ tBuf)           // [B_*NF]
{
    const int idx = blockIdx.x * blockDim.x + threadIdx.x;
    if (idx >= B_ * NF) return;
    const int b = idx / NF;
    const int f = idx % NF;

    float s = 0.f, q = 0.f;
    for (int m = 0; m < MT_PER_B; ++m) {
        const size_t p = (((size_t)b * MT_PER_B + m) * NF + f) * 2;
        s += part[p];
        q += part[p + 1];
    }
    const float mu  = s * (1.0f / HW);
    float var = q * (1.0f / HW) - mu * mu;
    var = (var < 0.f) ? 0.f : var;
    const float rsig = rsqrtf(var + 1e-8f);

    float g  = db[f];
    float bt = db[NF + f];
    const float* dl = dlat + b * C_;
    for (int z = 0; z < C_; ++z) {
        const float d = dl[z];
        g  += d * dw[(size_t)z * (2 * NF) + f];
        bt += d * dw[(size_t)z * (2 * NF) + NF + f];
    }
    const float sc = rsig * (1.f + g);
    sBuf[idx] = sc;
    tBuf[idx] = bt - mu * sc;
}

// ---------------------------------------------------------------------------
// Kernel 3: in-place normalize+affine, float4 (bandwidth bound, ~128 MB).
// ---------------------------------------------------------------------------
__global__ __launch_bounds__(256) void k3_norm(
    float* __restrict__ out,
    const float* __restrict__ sBuf,
    const float* __restrict__ tBuf)
{
    const size_t i4  = (size_t)blockIdx.x * blockDim.x + threadIdx.x;
    const size_t row = i4 >> 7;          // / (NF/4)
    const int b  = (int)(row >> 12);     // / HW
    const int c4 = (int)(i4 & 127);

    const float4 o  = ((const float4*)out)[i4];
    const float4 sv = ((const float4*)(sBuf + b * NF))[c4];
    const float4 tv = ((const float4*)(tBuf + b * NF))[c4];
    float4 r;
    r.x = o.x * sv.x + tv.x;
    r.y = o.y * sv.y + tv.y;
    r.z = o.z * sv.z + tv.z;
    r.w = o.w * sv.w + tv.w;
    ((float4*)out)[i4] = r;
}

// ---------------------------------------------------------------------------
extern "C" void kernel_launch(void* const* d_in, const int* in_sizes, int n_in,
                              void* d_out, int out_size, void* d_ws, size_t ws_size,
                              hipStream_t stream) {
    const float* x     = (const float*)d_in[0];
    const float* dlat  = (const float*)d_in[1];
    const float* noise = (const float*)d_in[2];
    const float* convw = (const float*)d_in[3];
    const float* nw    = (const float*)d_in[4];
    const float* bias  = (const float*)d_in[5];
    const float* dw    = (const float*)d_in[6];
    const float* db    = (const float*)d_in[7];
    float* out = (float*)d_out;

    float* part = (float*)d_ws;                               // 1 MB
    float* sBuf = part + (size_t)B_ * MT_PER_B * NF * 2;      // 16 KB
    float* tBuf = sBuf + (size_t)B_ * NF;                     // 16 KB

    dim3 g1(MTOT / BM, NF / BN);
    k1_gemm_act_stats<<<g1, 256, 0, stream>>>(x, convw, noise, nw, bias, out, part);

    k2_stats_style<<<(B_ * NF + 255) / 256, 256, 0, stream>>>(part, dlat, dw, db, sBuf, tBuf);

    const size_t n4 = (size_t)MTOT * NF / 4;                  // 4,194,304
    k3_norm<<<(unsigned)(n4 / 256), 256, 0, stream>>>(out, sBuf, tBuf);
}